// BinarySVM_70789650972758
// MI455X (gfx1250) — compile-verified
//
#include <hip/hip_runtime.h>

typedef __attribute__((ext_vector_type(2))) float v2f;
typedef __attribute__((ext_vector_type(4))) float v4f;
typedef __attribute__((ext_vector_type(8))) float v8f;

#define FDIM       64
#define NX         16384
#define NS         8192
#define GAMMA      (1.0f / 64.0f)
#define TWO_GAMMA  (1.0f / 32.0f)
#define SPLIT      4            // support-range split per x-tile (occupancy)

// ---- Pass 1: beta[j] = alpha[j] * exp(-gamma * ||s_j||^2)  (one-off, 8192 rows)
__global__ __launch_bounds__(256) void beta_kernel(
    const float* __restrict__ s,
    const float* __restrict__ alphas,
    float* __restrict__ beta)
{
    int j = blockIdx.x * blockDim.x + threadIdx.x;   // 0..NS-1
    const v4f* row = (const v4f*)(s + (size_t)j * FDIM);
    float ss = 0.0f;
#pragma unroll
    for (int i = 0; i < FDIM / 4; ++i) {
        v4f v = row[i];
        ss = __builtin_fmaf(v.x, v.x, ss);
        ss = __builtin_fmaf(v.y, v.y, ss);
        ss = __builtin_fmaf(v.z, v.z, ss);
        ss = __builtin_fmaf(v.w, v.w, ss);
    }
    beta[j] = alphas[j] * __expf(-GAMMA * ss);
}

// ---- Pass 2: per-wave 16-row x-tile vs. a 1/SPLIT slice of supports ----------
__global__ __launch_bounds__(256) void rbf_svm_wmma(
    const float* __restrict__ x,
    const float* __restrict__ s,
    const float* __restrict__ beta,
    float* __restrict__ partial)     // [SPLIT][NX]
{
    const int lane = threadIdx.x & 31;
    const int wave = (blockIdx.x * (blockDim.x >> 5)) + (threadIdx.x >> 5);
    const int tile = wave >> 2;            // 0..1023
    const int part = wave & (SPLIT - 1);   // 0..3
    const int m0   = tile * 16;
    const int l16  = lane & 15;
    const int half = lane >> 4;            // 0: K-pair {4k,4k+1}, 1: {4k+2,4k+3}

    const int nBeg = part * (NS / SPLIT);
    const int nEnd = nBeg + (NS / SPLIT);

    // A fragments for all 16 K-steps: x tile lives in VGPRs for the whole kernel
    v2f a[16];
    float xsq_part = 0.0f;
    const float* xrow = x + (size_t)(m0 + l16) * FDIM + 2 * half;
#pragma unroll
    for (int k = 0; k < 16; ++k) {
        v2f t = *(const v2f*)(xrow + 4 * k);
        a[k] = t;
        xsq_part = __builtin_fmaf(t.x, t.x, xsq_part);
        xsq_part = __builtin_fmaf(t.y, t.y, xsq_part);
    }
    float xsq_row = xsq_part + __shfl_xor(xsq_part, 16, 32);
    float xsq[8];                                   // xsq of row (r + 8*half)
#pragma unroll
    for (int r = 0; r < 8; ++r)
        xsq[r] = __shfl(xsq_row, r + 8 * half, 32);

    float acc[8];
#pragma unroll
    for (int r = 0; r < 8; ++r) acc[r] = 0.0f;

    for (int n0 = nBeg; n0 < nEnd; n0 += 32) {
        const float* brow0 = s + (size_t)(n0 + l16) * FDIM + 2 * half;
        const float* brow1 = brow0 + 16 * FDIM;
        if (n0 + 32 < nEnd)
            __builtin_prefetch(brow0 + 32 * FDIM, 0, 0);   // global_prefetch_b8

        v8f c0 = {}, c1 = {};
#pragma unroll
        for (int k = 0; k < 16; ++k) {
            v2f b0 = *(const v2f*)(brow0 + 4 * k);   // feeds WMMA directly
            v2f b1 = *(const v2f*)(brow1 + 4 * k);
            c0 = __builtin_amdgcn_wmma_f32_16x16x4_f32(
                     false, a[k], false, b0, (short)0, c0, false, false);
            c1 = __builtin_amdgcn_wmma_f32_16x16x4_f32(
                     false, a[k], false, b1, (short)0, c1, false, false);
        }
        float be0 = beta[n0 + l16];
        float be1 = beta[n0 + 16 + l16];
#pragma unroll
        for (int r = 0; r < 8; ++r) {
            acc[r] = __builtin_fmaf(be0, __expf(TWO_GAMMA * c0[r]), acc[r]);
            acc[r] = __builtin_fmaf(be1, __expf(TWO_GAMMA * c1[r]), acc[r]);
        }
    }

    // reduce over the 16 columns within each half-wave
#pragma unroll
    for (int r = 0; r < 8; ++r) {
        float v = acc[r];
        v += __shfl_xor(v, 1, 32);
        v += __shfl_xor(v, 2, 32);
        v += __shfl_xor(v, 4, 32);
        v += __shfl_xor(v, 8, 32);
        acc[r] = v;
    }
    if (l16 == 0) {
#pragma unroll
        for (int r = 0; r < 8; ++r)
            partial[(size_t)part * NX + m0 + 8 * half + r] =
                acc[r] * __expf(-GAMMA * xsq[r]);   // fold in exp(-g*||x||^2)
    }
}

// ---- Pass 3: deterministic fixed-order combine of the SPLIT partials --------
__global__ __launch_bounds__(256) void combine_kernel(
    const float* __restrict__ partial, float* __restrict__ out)
{
    int i = blockIdx.x * blockDim.x + threadIdx.x;   // 0..NX-1
    float v = partial[i];
#pragma unroll
    for (int p = 1; p < SPLIT; ++p)
        v += partial[(size_t)p * NX + i];
    out[i] = v;
}

extern "C" void kernel_launch(void* const* d_in, const int* in_sizes, int n_in,
                              void* d_out, int out_size, void* d_ws, size_t ws_size,
                              hipStream_t stream) {
    const float* x      = (const float*)d_in[0];
    const float* s      = (const float*)d_in[1];
    const float* alphas = (const float*)d_in[2];
    float* out          = (float*)d_out;

    float* beta    = (float*)d_ws;           // NS floats      (32 KB)
    float* partial = beta + NS;              // SPLIT*NX floats (256 KB)

    beta_kernel<<<NS / 256, 256, 0, stream>>>(s, alphas, beta);

    // 1024 tiles * SPLIT waves, 8 waves per block -> 512 blocks
    rbf_svm_wmma<<<(NX / 16) * SPLIT / 8, 256, 0, stream>>>(x, s, beta, partial);

    combine_kernel<<<NX / 256, 256, 0, stream>>>(partial, out);
}